// LGN_21852793602105
// MI455X (gfx1250) — compile-verified
//
#include <hip/hip_runtime.h>
#include <hip/hip_bf16.h>

typedef __attribute__((ext_vector_type(2))) float v2f;
typedef __attribute__((ext_vector_type(8))) float v8f;

#define D_DIM 64
#define N_USER_CONST 50000

// ---------------- degree / norm ----------------

__global__ void lgn_zero_f32(float* __restrict__ p, int n) {
    int i = blockIdx.x * blockDim.x + threadIdx.x;
    if (i < n) p[i] = 0.0f;
}

__global__ void lgn_zero_f32_vec4(float4* __restrict__ p, int n4) {
    int i = blockIdx.x * blockDim.x + threadIdx.x;
    if (i < n4) p[i] = make_float4(0.f, 0.f, 0.f, 0.f);
}

__global__ void lgn_copy_vec4(const float4* __restrict__ src, float4* __restrict__ dst, int n4) {
    int i = blockIdx.x * blockDim.x + threadIdx.x;
    if (i < n4) dst[i] = src[i];
}

__global__ void lgn_deg_count(const int* __restrict__ dst, float* __restrict__ deg, int E) {
    int e = blockIdx.x * blockDim.x + threadIdx.x;
    if (e < E) atomicAdd(&deg[dst[e]], 1.0f);
}

__global__ void lgn_norm(float* __restrict__ deg_to_norm, int n) {
    int i = blockIdx.x * blockDim.x + threadIdx.x;
    if (i < n) {
        float d = deg_to_norm[i];
        deg_to_norm[i] = rsqrtf(fmaxf(d, 1.0f));
    }
}

// ---------------- one LGConv layer ----------------
// scatter: hout[dst] += h[src] * norm[src]   (16 threads per edge, float4 each)
__global__ void lgn_scatter(const float* __restrict__ h,
                            const float* __restrict__ norm,
                            const int* __restrict__ src,
                            const int* __restrict__ dst,
                            float* __restrict__ hout, int E) {
    int t = blockIdx.x * blockDim.x + threadIdx.x;
    int e = t >> 4;
    if (e >= E) return;
    int c = (t & 15) * 4;
    int s = src[e];
    int d = dst[e];
    float ns = norm[s];
    const float4 v = *(const float4*)(h + (size_t)s * D_DIM + c);
    float* o = hout + (size_t)d * D_DIM + c;
    atomicAdd(o + 0, v.x * ns);
    atomicAdd(o + 1, v.y * ns);
    atomicAdd(o + 2, v.z * ns);
    atomicAdd(o + 3, v.w * ns);
}

// finalize: h = h*norm[node]; acc += h
__global__ void lgn_finalize(float* __restrict__ hbuf,
                             const float* __restrict__ norm,
                             float* __restrict__ acc, int ND) {
    int i = blockIdx.x * blockDim.x + threadIdx.x;
    if (i >= ND) return;
    float v = hbuf[i] * norm[i >> 6];  // D = 64
    hbuf[i] = v;
    acc[i] += v;
}

// ---------------- scoring (WMMA) ----------------
// One wave handles 16 (user, item_p, item_n) triples. For each K-chunk of 4:
//   A  = u[16 x 4]      (lane r<16: K=0,1 ; lane r+16: K=2,3 of row r)
//   Bp = ip^T [4 x 16]  (same per-lane addressing: lane holds column r)
// C accumulates u * ip^T over all 16 chunks; diagonal = per-triple dot products.
// features = acc/(L+1); the 1/4 on each operand folds into 1/16 on the score.
__global__ void lgn_score_wmma(const float* __restrict__ acc,
                               const int* __restrict__ user,
                               const int* __restrict__ item_p,
                               const int* __restrict__ item_n,
                               float* __restrict__ out, int B, int n_user) {
    int wave = (blockIdx.x * blockDim.x + threadIdx.x) >> 5;
    int lane = threadIdx.x & 31;
    int row0 = wave * 16;
    if (row0 >= B) return;  // uniform across wave

    int r  = lane & 15;
    int hi = lane >> 4;

    int u_idx = user[row0 + r];
    int p_idx = item_p[row0 + r] + n_user;
    int n_idx = item_n[row0 + r] + n_user;

    const float* uptr = acc + (size_t)u_idx * D_DIM;
    const float* pptr = acc + (size_t)p_idx * D_DIM;
    const float* nptr = acc + (size_t)n_idx * D_DIM;

    const float inv_sq = 1.0f / 16.0f;  // (1/(L+1))^2, L=3

#if __has_builtin(__builtin_amdgcn_wmma_f32_16x16x4_f32)
    v8f cp = {};
    v8f cn = {};
#pragma unroll
    for (int k = 0; k < 16; ++k) {
        int col = k * 4 + hi * 2;
        v2f a  = *(const v2f*)(uptr + col);
        v2f bp = *(const v2f*)(pptr + col);
        v2f bn = *(const v2f*)(nptr + col);
        cp = __builtin_amdgcn_wmma_f32_16x16x4_f32(false, a, false, bp, (short)0, cp, false, false);
        cn = __builtin_amdgcn_wmma_f32_16x16x4_f32(false, a, false, bn, (short)0, cn, false, false);
    }
    // C/D layout (16x16 f32): VGPR v holds M=v (lanes 0-15, N=lane) and M=v+8
    // (lanes 16-31, N=lane-16). Diagonal M==N:
    //   m = 0..7  -> lane m,      element m
    //   m = 8..15 -> lane m+16,   element m-8
    bool lo = (lane < 8);
    bool hi8 = (lane >= 24);
    int idx = lo ? lane : (lane - 24);
    float pv = cp[0], nv = cn[0];
#pragma unroll
    for (int rr = 1; rr < 8; ++rr) {
        pv = (idx == rr) ? cp[rr] : pv;
        nv = (idx == rr) ? cn[rr] : nv;
    }
    if (lo || hi8) {
        int m = lo ? lane : (lane - 16);
        out[row0 + m]     = pv * inv_sq;
        out[B + row0 + m] = nv * inv_sq;
    }
#else
    // Scalar fallback (should not be taken on gfx1250)
    if (hi == 0) {
        float ps = 0.f, ns = 0.f;
        for (int d = 0; d < D_DIM; ++d) {
            float uv = uptr[d];
            ps += uv * pptr[d];
            ns += uv * nptr[d];
        }
        out[row0 + r]     = ps * inv_sq;
        out[B + row0 + r] = ns * inv_sq;
    }
#endif
}

// ---------------- launch ----------------

extern "C" void kernel_launch(void* const* d_in, const int* in_sizes, int n_in,
                              void* d_out, int out_size, void* d_ws, size_t ws_size,
                              hipStream_t stream) {
    const float* emb    = (const float*)d_in[0];
    const int*   src    = (const int*)d_in[1];
    const int*   dst    = (const int*)d_in[2];
    const int*   user   = (const int*)d_in[3];
    const int*   item_p = (const int*)d_in[4];
    const int*   item_n = (const int*)d_in[5];

    const int ND = in_sizes[0];       // N * 64
    const int N  = ND / D_DIM;        // 100000
    const int E  = in_sizes[1];       // 1200000
    const int B  = in_sizes[3];       // 4096

    float* ws   = (float*)d_ws;
    float* norm = ws;                 // [N]   (deg, then norm in place)
    float* bufA = ws + N;             // [N*D]
    float* bufB = bufA + ND;          // [N*D]
    float* acc  = bufB + ND;          // [N*D] running layer sum

    const int TB = 256;
    const int ND4 = ND / 4;

    // degree -> norm
    lgn_zero_f32<<<(N + TB - 1) / TB, TB, 0, stream>>>(norm, N);
    lgn_deg_count<<<(E + TB - 1) / TB, TB, 0, stream>>>(dst, norm, E);
    lgn_norm<<<(N + TB - 1) / TB, TB, 0, stream>>>(norm, N);

    // acc = feats[0] = embeddings
    lgn_copy_vec4<<<(ND4 + TB - 1) / TB, TB, 0, stream>>>((const float4*)emb, (float4*)acc, ND4);

    // 3 LGConv layers, ping-pong bufA/bufB
    const float* cur = emb;
    float* pp[2] = {bufA, bufB};
    for (int l = 0; l < 3; ++l) {
        float* nxt = pp[l & 1];
        lgn_zero_f32_vec4<<<(ND4 + TB - 1) / TB, TB, 0, stream>>>((float4*)nxt, ND4);
        long long tot = (long long)E * 16;
        lgn_scatter<<<(int)((tot + TB - 1) / TB), TB, 0, stream>>>(cur, norm, src, dst, nxt, E);
        lgn_finalize<<<(ND + TB - 1) / TB, TB, 0, stream>>>(nxt, norm, acc, ND);
        cur = nxt;
    }

    // scoring: one wave per 16 triples -> B/16 waves -> B*2 threads
    int score_threads = (B / 16) * 32;
    lgn_score_wmma<<<(score_threads + TB - 1) / TB, TB, 0, stream>>>(
        acc, user, item_p, item_n, (float*)d_out, B, N_USER_CONST);
}